// LossYOLOv3_49864570307088
// MI455X (gfx1250) — compile-verified
//
#include <hip/hip_runtime.h>
#include <stdint.h>

#define NBX 50
#define NCLS 20
#define A_TOT 10647
#define A_STRIDE 10656
#define BATCH 32
#define TILE 256
#define NT2 1024
#define NT3 1024
#define THR_NEG 0.5f
#define NUM_NEG 1000

// ---------------- CDNA5 async global->LDS helpers ----------------
__device__ __forceinline__ void async_b128(uint32_t lds_off, const void* g) {
  asm volatile("global_load_async_to_lds_b128 %0, %1, off"
               :: "v"(lds_off), "v"((unsigned long long)(uintptr_t)g)
               : "memory");
}
__device__ __forceinline__ void wait_async0() {
#if __has_builtin(__builtin_amdgcn_s_wait_asynccnt)
  __builtin_amdgcn_s_wait_asynccnt(0);
#else
  asm volatile("s_wait_asynccnt 0x0" ::: "memory");
#endif
}

// ---------------- math helpers ----------------
__device__ __forceinline__ float fsig(float x) { return 1.f / (1.f + __expf(-x)); }
// log(sigmoid(x)), numerically stable
__device__ __forceinline__ float flogsig(float x) {
  return fminf(x, 0.f) - log1pf(__expf(-fabsf(x)));
}

// CIoU on ltrb boxes; atan(w/(h+eps)) precomputed per box (pair-invariant hoist)
__device__ __forceinline__ float ciou_pair(
    float l1, float t1, float r1, float b1, float w1, float h1, float at1,
    float l2, float t2, float r2, float b2, float w2, float h2, float at2) {
  const float eps = 1e-7f;
  float iw = fmaxf(fminf(r1, r2) - fmaxf(l1, l2), 0.f);
  float ih = fmaxf(fminf(b1, b2) - fmaxf(t1, t2), 0.f);
  float inter = iw * ih;
  float uni = w1 * h1 + w2 * h2 - inter + eps;
  float iou = inter / uni;
  float cw = fmaxf(r1, r2) - fminf(l1, l2);
  float ch = fmaxf(b1, b2) - fminf(t1, t2);
  float c2 = cw * cw + ch * ch + eps;
  float dx = (l1 + r1) - (l2 + r2);
  float dy = (t1 + b1) - (t2 + b2);
  float rho2 = 0.25f * (dx * dx + dy * dy);
  float dat = at1 - at2;
  float v = 0.405284734569351f * dat * dat;  // 4/pi^2
  float alpha = v / (1.f - iou + v + eps);
  return iou - rho2 / c2 - alpha * v;
}

__device__ __forceinline__ uint32_t ordered_key(float f) {
  uint32_t u = __float_as_uint(f);
  return (u & 0x80000000u) ? ~u : (u | 0x80000000u);
}

// ---------------- K0: zero output accumulator ----------------
__global__ void k_zero(float* o) { o[0] = 0.f; }

// ---------------- K1: per-anchor max CIoU + focal-neg values ----------------
__global__ __launch_bounds__(TILE) void k_neg_scores(
    const float* __restrict__ p, const float* __restrict__ boxes,
    const float* __restrict__ ancs, uint32_t* __restrict__ key,
    float* __restrict__ fneg) {
  __shared__ __align__(16) float sp[TILE * 25 + 8];  // staged p tile (+ misalign slack)
  __shared__ __align__(16) float sanc[TILE * 4];
  __shared__ float sbox[NBX * 8];  // l,t,r,b,w,h,atan,(pad)
  const int b = blockIdx.y;
  const int a0 = blockIdx.x * TILE;
  const int cnt = min(TILE, A_TOT - a0);
  const int tid = threadIdx.x;

  // async-stage p[b, a0:a0+cnt, 0:25] (row = 100 B, base only 4B-aligned)
  uintptr_t gaddr = (uintptr_t)p + ((size_t)b * A_TOT + (size_t)a0) * 100u;
  uint32_t delta = (uint32_t)(gaddr & 15u);
  uintptr_t gal = gaddr - delta;
  int chunks = (cnt * 100 + (int)delta + 15) >> 4;
  uint32_t lds_p = (uint32_t)(uintptr_t)sp;
  for (int c = tid; c < chunks; c += TILE)
    async_b128(lds_p + (uint32_t)c * 16u, (const void*)(gal + (size_t)c * 16u));
  // async-stage ancs[a0:a0+cnt, 0:4] (16 B rows, aligned)
  uintptr_t ga = (uintptr_t)ancs + (size_t)a0 * 16u;
  uint32_t lds_a = (uint32_t)(uintptr_t)sanc;
  for (int c = tid; c < cnt; c += TILE)
    async_b128(lds_a + (uint32_t)c * 16u, (const void*)(ga + (size_t)c * 16u));

  if (tid < NBX) {
    const float* bp = boxes + ((size_t)b * NBX + tid) * 4;
    float l = bp[0], t = bp[1], r = bp[2], bb = bp[3];
    float w = r - l, h = bb - t;
    float* s = &sbox[tid * 8];
    s[0] = l; s[1] = t; s[2] = r; s[3] = bb;
    s[4] = w; s[5] = h; s[6] = atanf(w / (h + 1e-7f));
  }
  wait_async0();
  __syncthreads();

  if (tid < cnt) {
    const int a = a0 + tid;
    const float fs = (a < 8112) ? 52.f : (a < 10140 ? 26.f : 13.f);
    const float* pr = &sp[(delta >> 2) + tid * 25];
    const float* an = &sanc[tid * 4];
    float px = fsig(pr[0]) / fs + an[0];
    float py = fsig(pr[1]) / fs + an[1];
    float pw = __expf(pr[2]) * an[2];
    float ph = __expf(pr[3]) * an[3];
    float conf = pr[4];
    float l2 = px - pw * 0.5f, t2 = py - ph * 0.5f;
    float r2 = px + pw * 0.5f, b2 = py + ph * 0.5f;
    float at2 = atanf(pw / (ph + 1e-7f));
    float mx = -INFINITY;
    #pragma unroll 5
    for (int j = 0; j < NBX; j++) {
      const float* s = &sbox[j * 8];
      float c = ciou_pair(s[0], s[1], s[2], s[3], s[4], s[5], s[6],
                          l2, t2, r2, b2, pw, ph, at2);
      mx = fmaxf(mx, c);
    }
    float masked = (mx < THR_NEG) ? mx : INFINITY;
    size_t o = (size_t)b * A_STRIDE + (size_t)a;
    key[o] = ordered_key(masked);
    float prb = fsig(conf);  // focal(x, 0) = 0.75 * pr^2 * (-log sigmoid(-x))
    fneg[o] = 0.75f * prb * prb * (-flogsig(-conf));
  }
}

// ---------------- K2: in-LDS radix select of 1000 smallest + focal-neg sum ----
__global__ __launch_bounds__(NT2) void k_select_neg(
    const uint32_t* __restrict__ key, const float* __restrict__ fneg,
    float* __restrict__ out) {
  __shared__ __align__(16) uint32_t skey[A_STRIDE];  // 42.6 KB (big-LDS win)
  __shared__ uint32_t hist[256];
  __shared__ uint32_t scan[NT2];
  __shared__ float red[NT2];
  __shared__ uint32_t sh_prefix, sh_rem;
  const int b = blockIdx.x, tid = threadIdx.x;

  // async-stage all keys for this image into LDS
  uintptr_t g = (uintptr_t)(key + (size_t)b * A_STRIDE);
  uint32_t lds = (uint32_t)(uintptr_t)skey;
  const int chunks = (A_TOT * 4 + 15) >> 4;
  for (int c = tid; c < chunks; c += NT2)
    async_b128(lds + (uint32_t)c * 16u, (const void*)(g + (size_t)c * 16u));
  if (tid == 0) { sh_prefix = 0u; sh_rem = NUM_NEG; }
  wait_async0();
  __syncthreads();

  // 4-pass MSB-first radix select of the NUM_NEG-th smallest key
  for (int shift = 24; shift >= 0; shift -= 8) {
    if (tid < 256) hist[tid] = 0u;
    __syncthreads();
    uint32_t pfx = sh_prefix;
    for (int i = tid; i < A_TOT; i += NT2) {
      uint32_t u = skey[i];
      bool in = (shift == 24) || ((u >> (shift + 8)) == pfx);
      if (in) atomicAdd(&hist[(u >> shift) & 255u], 1u);
    }
    __syncthreads();
    if (tid == 0) {
      uint32_t cum = 0, d = 0, rem = sh_rem;
      for (d = 0; d < 256; d++) {
        uint32_t h = hist[d];
        if (cum + h >= rem) break;
        cum += h;
      }
      sh_rem = rem - cum;
      sh_prefix = (pfx << 8) | d;
    }
    __syncthreads();
  }
  const uint32_t K = sh_prefix;
  const uint32_t rem = sh_rem;  // # of ties at K to take (lowest index first)

  const float* fn = fneg + (size_t)b * A_STRIDE;
  const int chunk = (A_TOT + NT2 - 1) / NT2;
  const int i0 = tid * chunk, i1 = min(i0 + chunk, A_TOT);
  float s = 0.f;
  uint32_t eq = 0;
  for (int i = i0; i < i1; i++) {
    uint32_t u = skey[i];
    if (u < K) s += fn[i];
    else if (u == K) eq++;
  }
  // block exclusive scan over per-thread tie counts (index-ordered tie-break)
  scan[tid] = eq;
  __syncthreads();
  for (int off = 1; off < NT2; off <<= 1) {
    uint32_t v = (tid >= off) ? scan[tid - off] : 0u;
    __syncthreads();
    scan[tid] += v;
    __syncthreads();
  }
  uint32_t rank = scan[tid] - eq;
  for (int i = i0; i < i1; i++) {
    if (skey[i] == K) { if (rank < rem) s += fn[i]; rank++; }
  }
  red[tid] = s;
  __syncthreads();
  for (int off = NT2 >> 1; off > 0; off >>= 1) {
    if (tid < off) red[tid] += red[tid + off];
    __syncthreads();
  }
  if (tid == 0) atomicAdd(out, red[0] * (1.f / BATCH));
}

// ---------------- K3: positives (matching + cls/iou/conf-pos losses) --------
__global__ __launch_bounds__(NT3) void k_pos(
    const float* __restrict__ p, const float* __restrict__ boxes,
    const int* __restrict__ labels, const float* __restrict__ ancs,
    float* __restrict__ out) {
  const float ASCW[9] = {10.f/416.f, 16.f/416.f, 33.f/416.f, 30.f/416.f, 62.f/416.f,
                         59.f/416.f, 116.f/416.f, 156.f/416.f, 373.f/416.f};
  const float ASCH[9] = {13.f/416.f, 30.f/416.f, 23.f/416.f, 61.f/416.f, 45.f/416.f,
                         119.f/416.f, 90.f/416.f, 198.f/416.f, 326.f/416.f};
  const float FSL[3] = {52.f, 26.f, 13.f};
  const int FSI[3] = {52, 26, 13};
  const int CUMC[3] = {0, 2704, 3380};

  __shared__ float sb[NBX * 10];  // l,t,r,b,cx,cy,w,h,atan,(pad)
  __shared__ float pv[NT3];
  __shared__ int pi[NT3];
  __shared__ int smidx[NBX];
  __shared__ float s_cls, s_iou, s_cpos;
  const int b = blockIdx.x, tid = threadIdx.x;

  if (tid < NBX) {
    const float* bp = boxes + ((size_t)b * NBX + tid) * 4;
    float l = bp[0], t = bp[1], r = bp[2], bo = bp[3];
    float* s = &sb[tid * 10];
    s[0] = l; s[1] = t; s[2] = r; s[3] = bo;
    s[4] = (l + r) * 0.5f; s[5] = (t + bo) * 0.5f;
    s[6] = r - l; s[7] = bo - t;
    s[8] = atanf((r - l) / ((bo - t) + 1e-7f));
  }
  if (tid == 0) { s_cls = 0.f; s_iou = 0.f; s_cpos = 0.f; }
  __syncthreads();

  // [nb, nb*9] CIoU with the index-offset trick; argmax with first-max tie-break
  const int row = tid >> 4, lane = tid & 15;
  float best = -INFINITY;
  int bidx = 0x7fffffff;
  if (row < NBX) {
    const float* s = &sb[row * 10];
    const float off1 = (float)row;
    for (int j = lane; j < NBX * 9; j += 16) {
      int jb = j / 9, jk = j % 9;
      const float* sj = &sb[jb * 10];
      float fs = FSL[jk / 3];
      float acx = floorf(sj[4] * fs) / fs;
      float acy = floorf(sj[5] * fs) / fs;
      float aw = ASCW[jk], ah = ASCH[jk];
      float off2 = (float)jb;
      float c = ciou_pair(s[0] + off1, s[1] + off1, s[2] + off1, s[3] + off1,
                          s[6], s[7], s[8],
                          acx - aw * 0.5f + off2, acy - ah * 0.5f + off2,
                          acx + aw * 0.5f + off2, acy + ah * 0.5f + off2,
                          aw, ah, atanf(aw / (ah + 1e-7f)));
      if (c > best) { best = c; bidx = j; }  // ascending j keeps lowest index
    }
  }
  pv[tid] = best;
  pi[tid] = bidx;
  __syncthreads();

  if (row < NBX && lane == 0) {
    float bv = pv[tid];
    int bi = pi[tid];
    for (int l2 = 1; l2 < 16; l2++) {
      float v = pv[tid + l2];
      int ii = pi[tid + l2];
      if (v > bv || (v == bv && ii < bi)) { bv = v; bi = ii; }
    }
    int k = bi % 9;
    int lvl = k / 3;
    int fsi = FSI[lvl];
    float fsf = FSL[lvl];
    const float* s = &sb[row * 10];
    int col = (int)floorf(s[4] * fsf);
    int rw  = (int)floorf(s[5] * fsf);
    int midx = (CUMC[lvl] + rw * fsi + col) * 3 + lvl;  // matches reference
    smidx[row] = midx;

    // decode prediction at midx, CIoU loss + focal-pos conf
    const float* pp = p + ((size_t)b * A_TOT + (size_t)midx) * 25;
    const float* an = ancs + (size_t)midx * 4;
    float px = fsig(pp[0]) / fsf + an[0];
    float py = fsig(pp[1]) / fsf + an[1];
    float pw = __expf(pp[2]) * an[2];
    float ph = __expf(pp[3]) * an[3];
    float c1 = ciou_pair(s[0], s[1], s[2], s[3], s[6], s[7], s[8],
                         px - pw * 0.5f, py - ph * 0.5f,
                         px + pw * 0.5f, py + ph * 0.5f,
                         pw, ph, atanf(pw / (ph + 1e-7f)));
    float w = 2.f - s[6] * s[7];
    atomicAdd(&s_iou, w * (1.f - c1));
    float x = pp[4];
    float d = 1.f - fsig(x);  // focal(x, 1) = 0.25 * (1-pr)^2 * (-log sigmoid(x))
    atomicAdd(&s_cpos, 0.25f * d * d * (-flogsig(x)));
  }
  __syncthreads();

  if (tid < NBX * NCLS) {
    int i = tid / NCLS, c = tid % NCLS;
    int midx = smidx[i];
    float x = p[((size_t)b * A_TOT + (size_t)midx) * 25 + 5 + c];
    int lab = labels[(size_t)b * NBX + i] - 1;
    float bce = (c == lab) ? -flogsig(x) : -flogsig(-x);
    atomicAdd(&s_cls, bce);
  }
  __syncthreads();
  if (tid == 0) {
    float tot = (s_cls + s_iou + s_cpos) * (1.f / (float)NBX);
    atomicAdd(out, tot * (1.f / BATCH));
  }
}

// ---------------- host launcher ----------------
extern "C" void kernel_launch(void* const* d_in, const int* in_sizes, int n_in,
                              void* d_out, int out_size, void* d_ws, size_t ws_size,
                              hipStream_t stream) {
  (void)in_sizes; (void)n_in; (void)out_size; (void)ws_size;
  const float* p      = (const float*)d_in[0];  // [B, A, 25]
  const float* boxes  = (const float*)d_in[1];  // [B, 50, 4] ltrb
  const int*   labels = (const int*)d_in[2];    // [B, 50]
  const float* ancs   = (const float*)d_in[3];  // [A, 4] xywh
  float* out = (float*)d_out;

  uint32_t* key = (uint32_t*)d_ws;
  float* fneg = (float*)((char*)d_ws + (size_t)BATCH * A_STRIDE * sizeof(uint32_t));

  k_zero<<<1, 1, 0, stream>>>(out);
  dim3 g1((A_TOT + TILE - 1) / TILE, BATCH);
  k_neg_scores<<<g1, TILE, 0, stream>>>(p, boxes, ancs, key, fneg);
  k_select_neg<<<BATCH, NT2, 0, stream>>>(key, fneg, out);
  k_pos<<<BATCH, NT3, 0, stream>>>(p, boxes, labels, ancs, out);
}